// BiMambaBlock_34866544508977
// MI455X (gfx1250) — compile-verified
//
#include <hip/hip_runtime.h>

// ---------------- problem constants (from reference) ----------------
#define B_SZ   2
#define L_SZ   1024
#define DM     512          // D_MODEL
#define DI     1024         // D_INNER
#define DSTATE 64
#define DTR    32           // DT_RANK
#define XDBL_N 160          // DT_RANK + 2*D_STATE
#define TOK    (B_SZ * L_SZ)

typedef float v2f __attribute__((ext_vector_type(2)));
typedef float v4f __attribute__((ext_vector_type(4)));
typedef float v8f __attribute__((ext_vector_type(8)));

__device__ __forceinline__ float sigmoidf_(float x) { return 1.f / (1.f + __expf(-x)); }
__device__ __forceinline__ float siluf_(float x)    { return x * sigmoidf_(x); }
__device__ __forceinline__ float softplusf_(float x){ return (x > 20.f) ? x : log1pf(__expf(x)); }

// ---------------- WMMA fp32 GEMM:  C[M,N] = A[M,K(lda)] * W[N,K]^T ----------------
// Block: 256 thr = 8 waves; block tile 32(M) x 128(N); wave tile 16x32
// (two 16x16 accumulators sharing one A fragment -> 2x A reuse, 16 WMMA per
// barrier pair). KC=32 staged through padded LDS.
// EPI: 0 none, 1 +bias, 2 +bias,softplus, 3 silu
#define KC   32
#define LDSP 36

template <int EPI>
__global__ __launch_bounds__(256) void gemm_wmma(
    const float* __restrict__ A, int lda,
    const float* __restrict__ Wt,          // [N,K] row-major (weights)
    const float* __restrict__ bias,        // [N] or nullptr
    float* __restrict__ Cout,              // [M,N]
    int M, int N, int K)
{
    __shared__ float As[32][LDSP];
    __shared__ float Bs[128][LDSP];

    const int tid  = threadIdx.x;
    const int lane = tid & 31;
    const int wave = tid >> 5;
    const int half = lane >> 4;     // 0/1
    const int lm   = lane & 15;
    const int m0   = blockIdx.y << 5;
    const int n0   = blockIdx.x << 7;
    const int wr   = (wave >> 2) << 4;   // 0 or 16
    const int wc   = (wave & 3) << 5;    // 0,32,64,96

    v8f acc0 = {};
    v8f acc1 = {};

    const int ar  = tid >> 3;            // 0..31
    const int ac4 = (tid & 7) << 2;      // 0..28

    for (int k0 = 0; k0 < K; k0 += KC) {
        // stage A tile 32xKC (one float4 per thread)
        v4f av = *(const v4f*)(A + (size_t)(m0 + ar) * lda + k0 + ac4);
        As[ar][ac4 + 0] = av.x; As[ar][ac4 + 1] = av.y;
        As[ar][ac4 + 2] = av.z; As[ar][ac4 + 3] = av.w;
        // stage B tile 128xKC (four float4 per thread), zero-pad past N
        #pragma unroll
        for (int i = 0; i < 4; ++i) {
            int t2  = tid + (i << 8);
            int br  = t2 >> 3;
            int bc4 = (t2 & 7) << 2;
            v4f bv = {0.f, 0.f, 0.f, 0.f};
            if (n0 + br < N)
                bv = *(const v4f*)(Wt + (size_t)(n0 + br) * K + k0 + bc4);
            Bs[br][bc4 + 0] = bv.x; Bs[br][bc4 + 1] = bv.y;
            Bs[br][bc4 + 2] = bv.z; Bs[br][bc4 + 3] = bv.w;
        }
        if (k0 + KC < K)
            __builtin_prefetch(A + (size_t)(m0 + ar) * lda + k0 + KC + ac4, 0, 3);
        __syncthreads();

        // 16 WMMA steps per chunk.  ISA f32 16x16x4 layouts:
        //   A: lanes0-15 hold M=lane,K={k,k+1}; lanes16-31 hold M=lane-16,K={k+2,k+3}
        //   B: lanes0-15 hold N=lane,K={k,k+1}; lanes16-31 hold N=lane-16,K={k+2,k+3}
        //   C: VGPR r: lanes0-15 M=r,N=lane; lanes16-31 M=8+r,N=lane-16
        #pragma unroll
        for (int kk = 0; kk < KC; kk += 4) {
            v2f a, b0, b1;
            a.x  = As[wr + lm][kk + (half << 1) + 0];
            a.y  = As[wr + lm][kk + (half << 1) + 1];
            b0.x = Bs[wc + lm][kk + (half << 1) + 0];
            b0.y = Bs[wc + lm][kk + (half << 1) + 1];
            b1.x = Bs[wc + 16 + lm][kk + (half << 1) + 0];
            b1.y = Bs[wc + 16 + lm][kk + (half << 1) + 1];
            acc0 = __builtin_amdgcn_wmma_f32_16x16x4_f32(
                false, a, false, b0, (short)0, acc0, false, false);
            acc1 = __builtin_amdgcn_wmma_f32_16x16x4_f32(
                false, a, false, b1, (short)0, acc1, false, false);
        }
        __syncthreads();
    }

    // epilogue: two 16x16 tiles per wave
    #pragma unroll
    for (int tile = 0; tile < 2; ++tile) {
        const int ncol = n0 + wc + (tile << 4) + lm;
        if (ncol < N) {
            float bvv = 0.f;
            if (EPI == 1 || EPI == 2) bvv = bias[ncol];
            #pragma unroll
            for (int r = 0; r < 8; ++r) {
                float v = (tile ? acc1[r] : acc0[r]) + bvv;
                if (EPI == 2) v = softplusf_(v);
                if (EPI == 3) v = siluf_(v);
                int mrow = m0 + wr + (half << 3) + r;
                Cout[(size_t)mrow * N + ncol] = v;
            }
        }
    }
}

// ---------------- RMSNorm (input) : one block per token, D=512 ----------------
__global__ __launch_bounds__(256) void rmsnorm_in_kernel(
    const float* __restrict__ x, const float* __restrict__ w, float* __restrict__ h)
{
    __shared__ float sh[256];
    const int token = blockIdx.x;
    const int tid   = threadIdx.x;
    const float* xr = x + (size_t)token * DM;
    float v0 = xr[tid], v1 = xr[tid + 256];
    sh[tid] = v0 * v0 + v1 * v1;
    __syncthreads();
    for (int o = 128; o > 0; o >>= 1) {
        if (tid < o) sh[tid] += sh[tid + o];
        __syncthreads();
    }
    float inv = rsqrtf(sh[0] * (1.f / DM) + 1e-6f);
    h[(size_t)token * DM + tid]       = w[tid]       * (v0 * inv);
    h[(size_t)token * DM + tid + 256] = w[tid + 256] * (v1 * inv);
}

// ---------------- residual + RMSNorm (output) ----------------
__global__ __launch_bounds__(256) void residual_rmsnorm_out_kernel(
    const float* __restrict__ x, const float* __restrict__ ff,
    const float* __restrict__ w, float* __restrict__ out)
{
    __shared__ float sh[256];
    const int token = blockIdx.x;
    const int tid   = threadIdx.x;
    const size_t base = (size_t)token * DM;
    float v0 = x[base + tid]       + ff[base + tid];
    float v1 = x[base + tid + 256] + ff[base + tid + 256];
    sh[tid] = v0 * v0 + v1 * v1;
    __syncthreads();
    for (int o = 128; o > 0; o >>= 1) {
        if (tid < o) sh[tid] += sh[tid + o];
        __syncthreads();
    }
    float inv = rsqrtf(sh[0] * (1.f / DM) + 1e-6f);
    out[base + tid]       = w[tid]       * (v0 * inv);
    out[base + tid + 256] = w[tid + 256] * (v1 * inv);
}

// ---------------- depthwise causal conv(4) + SiLU, fwd + flipped ----------------
// dir==1 works in tau coordinates: u_f[tau] = u[b, L-1-tau]
__global__ __launch_bounds__(256) void conv_silu_kernel(
    const float* __restrict__ xz, const float* __restrict__ cw,
    const float* __restrict__ cb, float* __restrict__ ua0, float* __restrict__ ua1)
{
    size_t idx = (size_t)blockIdx.x * 256 + threadIdx.x;
    if (idx >= (size_t)2 * B_SZ * L_SZ * DI) return;
    int d = (int)(idx & (DI - 1));
    size_t r = idx >> 10;                 // / DI
    int t = (int)(r % L_SZ);  r /= L_SZ;
    int b = (int)(r % B_SZ);
    int dir = (int)(r / B_SZ);
    float acc = cb[d];
    #pragma unroll
    for (int j = 0; j < 4; ++j) {
        int i = t - 3 + j;
        if (i >= 0) {
            int torig = dir ? (L_SZ - 1 - i) : i;
            acc += cw[d * 4 + j] * xz[((size_t)(b * L_SZ + torig)) * (2 * DI) + d];
        }
    }
    float* dst = dir ? ua1 : ua0;
    dst[((size_t)(b * L_SZ) + t) * DI + d] = siluf_(acc);
}

// ---------------- selective scan: thread = (b,dir,channel), 64-state in regs ----------------
__global__ __launch_bounds__(256) void scan_kernel(
    const float* __restrict__ xz,     // [B,L,2048] (z at +DI), original time order
    const float* __restrict__ ua,     // [B,L,DI]   this dir (tau order)
    const float* __restrict__ xdbl,   // [B,L,160]  this dir (tau order)
    const float* __restrict__ dt,     // [B,L,DI]   this dir (tau order)
    const float* __restrict__ A_log,  // [DI,64]
    const float* __restrict__ Dskip,  // [DI]
    float* __restrict__ yg,           // [B,L,DI] gated output (tau order)
    int dir)
{
    const int d = blockIdx.x * 256 + threadIdx.x;   // grid.x = DI/256
    const int b = blockIdx.y;
    __shared__ float sB[DSTATE];
    __shared__ float sC[DSTATE];

    float s[DSTATE];
    float Ar[DSTATE];
    #pragma unroll
    for (int n = 0; n < DSTATE; ++n) {
        s[n]  = 0.f;
        Ar[n] = -__expf(A_log[(size_t)d * DSTATE + n]);
    }
    const float Dd = Dskip[d];

    for (int tau = 0; tau < L_SZ; ++tau) {
        const size_t row = (size_t)b * L_SZ + tau;
        __syncthreads();
        if (threadIdx.x < DSTATE)
            sB[threadIdx.x] = xdbl[row * XDBL_N + DTR + threadIdx.x];
        else if (threadIdx.x < 2 * DSTATE)
            sC[threadIdx.x - DSTATE] = xdbl[row * XDBL_N + DTR + DSTATE + (threadIdx.x - DSTATE)];
        __syncthreads();

        const float dtv = dt[row * DI + d];
        const float uv  = ua[row * DI + d];
        const float du  = dtv * uv;
        float y = 0.f;
        #pragma unroll
        for (int n = 0; n < DSTATE; ++n) {
            float dA = __expf(dtv * Ar[n]);
            s[n] = s[n] * dA + du * sB[n];
            y += s[n] * sC[n];
        }
        const int torig = dir ? (L_SZ - 1 - tau) : tau;
        const float z = xz[((size_t)(b * L_SZ + torig)) * (2 * DI) + DI + d];
        yg[row * DI + d] = (y + uv * Dd) * siluf_(z);
    }
}

// ---------------- hcat assembly (flip backward direction back to time order) ----------------
__global__ __launch_bounds__(256) void assemble_hcat_kernel(
    const float* __restrict__ hf, const float* __restrict__ hb, float* __restrict__ hcat)
{
    size_t idx = (size_t)blockIdx.x * 256 + threadIdx.x;
    if (idx >= (size_t)TOK * DM) return;
    size_t token = idx / DM;
    int c = (int)(idx % DM);
    int t = (int)(token % L_SZ);
    int b = (int)(token / L_SZ);
    hcat[token * (2 * DM) + c] = hf[token * DM + c];
    size_t tokb = (size_t)b * L_SZ + (L_SZ - 1 - t);
    hcat[token * (2 * DM) + DM + c] = hb[tokb * DM + c];
}

// ---------------- GLU gate + SiLU (FF input) ----------------
__global__ __launch_bounds__(256) void glu_silu_kernel(
    const float* __restrict__ uv, float* __restrict__ hact)
{
    size_t idx = (size_t)blockIdx.x * 256 + threadIdx.x;
    if (idx >= (size_t)TOK * DM) return;
    size_t row = idx / DM;
    int c = (int)(idx % DM);
    float ug = uv[row * (2 * DM) + c];
    float vg = uv[row * (2 * DM) + DM + c];
    float hmid = ug * sigmoidf_(vg);
    hact[idx] = siluf_(hmid);
}

// ---------------- launch ----------------
extern "C" void kernel_launch(void* const* d_in, const int* in_sizes, int n_in,
                              void* d_out, int out_size, void* d_ws, size_t ws_size,
                              hipStream_t stream)
{
    const float* x         = (const float*)d_in[0];
    const float* W_in      = (const float*)d_in[1];
    const float* conv_w    = (const float*)d_in[2];
    const float* conv_b    = (const float*)d_in[3];
    const float* W_xproj   = (const float*)d_in[4];
    const float* W_dt      = (const float*)d_in[5];
    const float* b_dt      = (const float*)d_in[6];
    const float* A_log     = (const float*)d_in[7];
    const float* Dskip     = (const float*)d_in[8];
    const float* W_out     = (const float*)d_in[9];
    const float* norm_in_w = (const float*)d_in[10];
    const float* fuse_W    = (const float*)d_in[11];
    const float* fuse_b    = (const float*)d_in[12];
    const float* ff_W1     = (const float*)d_in[13];
    const float* ff_W2     = (const float*)d_in[14];
    const float* norm_out_w= (const float*)d_in[15];
    float* out = (float*)d_out;

    // workspace layout (floats), with lifetime-based reuse (~74 MB total)
    float* wsf = (float*)d_ws;
    size_t off = 0;
    auto take = [&](size_t n) { float* p = wsf + off; off += n; return p; };
    float* h    = take((size_t)TOK * DM);       // later: hact
    float* xz   = take((size_t)TOK * 2 * DI);   // later: ffa
    float* ua0  = take((size_t)TOK * DI);       // later: ff2
    float* ua1  = take((size_t)TOK * DI);
    float* xd0  = take((size_t)TOK * XDBL_N);
    float* xd1  = take((size_t)TOK * XDBL_N);
    float* dt0  = take((size_t)TOK * DI);       // later: uv
    float* dt1  = take((size_t)TOK * DI);       // later: hf|hb
    float* yg0  = take((size_t)TOK * DI);       // later: hcat
    float* yg1  = take((size_t)TOK * DI);
    float* hact = h;
    float* ffa  = xz;
    float* ff2  = ua0;
    float* uv   = dt0;
    float* hf   = dt1;
    float* hb   = dt1 + (size_t)TOK * DM;
    float* hcat = yg0;

    const dim3 blk(256);

    // 1) RMSNorm in
    rmsnorm_in_kernel<<<TOK, blk, 0, stream>>>(x, norm_in_w, h);

    // 2) xz = h @ W_in^T   [2048 x 2048, K=512]
    { dim3 g((2 * DI + 127) / 128, TOK / 32);
      gemm_wmma<0><<<g, blk, 0, stream>>>(h, DM, W_in, nullptr, xz, TOK, 2 * DI, DM); }

    // 3) depthwise conv + SiLU (both directions)
    { size_t tot = (size_t)2 * B_SZ * L_SZ * DI;
      conv_silu_kernel<<<(unsigned)((tot + 255) / 256), blk, 0, stream>>>(xz, conv_w, conv_b, ua0, ua1); }

    // 4) x_dbl = u @ W_xproj^T   [2048 x 160, K=1024]
    { dim3 g((XDBL_N + 127) / 128, TOK / 32);
      gemm_wmma<0><<<g, blk, 0, stream>>>(ua0, DI, W_xproj, nullptr, xd0, TOK, XDBL_N, DI);
      gemm_wmma<0><<<g, blk, 0, stream>>>(ua1, DI, W_xproj, nullptr, xd1, TOK, XDBL_N, DI); }

    // 5) dt = softplus(x_dbl[:, :32] @ W_dt^T + b_dt)   [2048 x 1024, K=32, lda=160]
    { dim3 g((DI + 127) / 128, TOK / 32);
      gemm_wmma<2><<<g, blk, 0, stream>>>(xd0, XDBL_N, W_dt, b_dt, dt0, TOK, DI, DTR);
      gemm_wmma<2><<<g, blk, 0, stream>>>(xd1, XDBL_N, W_dt, b_dt, dt1, TOK, DI, DTR); }

    // 6) selective scans (fused Dskip + SiLU(z) gate)
    { dim3 g(DI / 256, B_SZ);
      scan_kernel<<<g, blk, 0, stream>>>(xz, ua0, xd0, dt0, A_log, Dskip, yg0, 0);
      scan_kernel<<<g, blk, 0, stream>>>(xz, ua1, xd1, dt1, A_log, Dskip, yg1, 1); }

    // 7) hf/hb = yg @ W_out^T   [2048 x 512, K=1024]
    { dim3 g((DM + 127) / 128, TOK / 32);
      gemm_wmma<0><<<g, blk, 0, stream>>>(yg0, DI, W_out, nullptr, hf, TOK, DM, DI);
      gemm_wmma<0><<<g, blk, 0, stream>>>(yg1, DI, W_out, nullptr, hb, TOK, DM, DI); }

    // 8) hcat (backward flipped back to time order)
    { size_t tot = (size_t)TOK * DM;
      assemble_hcat_kernel<<<(unsigned)((tot + 255) / 256), blk, 0, stream>>>(hf, hb, hcat); }

    // 9) uv = hcat @ fuse_W^T + fuse_b   [2048 x 1024, K=1024]
    { dim3 g((2 * DM + 127) / 128, TOK / 32);
      gemm_wmma<1><<<g, blk, 0, stream>>>(hcat, 2 * DM, fuse_W, fuse_b, uv, TOK, 2 * DM, 2 * DM); }

    // 10) hact = silu(ug * sigmoid(vg))
    { size_t tot = (size_t)TOK * DM;
      glu_silu_kernel<<<(unsigned)((tot + 255) / 256), blk, 0, stream>>>(uv, hact); }

    // 11) ffa = silu(hact @ ff_W1^T)   [2048 x 2048, K=512], silu in epilogue
    { dim3 g((4 * DM + 127) / 128, TOK / 32);
      gemm_wmma<3><<<g, blk, 0, stream>>>(hact, DM, ff_W1, nullptr, ffa, TOK, 4 * DM, DM); }

    // 12) ff2 = ffa @ ff_W2^T   [2048 x 512, K=2048]
    { dim3 g((DM + 127) / 128, TOK / 32);
      gemm_wmma<0><<<g, blk, 0, stream>>>(ffa, 4 * DM, ff_W2, nullptr, ff2, TOK, DM, 4 * DM); }

    // 13) out = rmsnorm(x + ff2)
    residual_rmsnorm_out_kernel<<<TOK, blk, 0, stream>>>(x, ff2, norm_out_w, out);
}